// SAE_83279415869641
// MI455X (gfx1250) — compile-verified
//
#include <hip/hip_runtime.h>

// SAE hierarchical quantizer, fused single-pass for MI455X (gfx1250, wave32).
// GEMM (65536x1024)x(1024x16pad) via V_WMMA_F32_16X16X4_F32, gumbel-argmax via
// wave shuffles, codebook row gather -> out. Bandwidth bound (~306 MB moved).

typedef float v2f __attribute__((ext_vector_type(2)));
typedef float v8f __attribute__((ext_vector_type(8)));

#define B_  128
#define N_  512
#define E_  1024
#define D_  64
// heads: widths 8,4,2 -> 14 logit columns, padded to 16 for WMMA

__device__ __forceinline__ float gumbel_from_u(float u) {
    u = fminf(fmaxf(u, 1e-10f), 1.0f - 1e-10f);
    return -__logf(-__logf(u));
}

__global__ __launch_bounds__(256)
void sae_quantize_kernel(const float* __restrict__ pt,
                         const float* __restrict__ g0,
                         const float* __restrict__ g1,
                         const float* __restrict__ g2,
                         const float* __restrict__ W0,
                         const float* __restrict__ b0,
                         const float* __restrict__ W1,
                         const float* __restrict__ b1,
                         const float* __restrict__ W2,
                         const float* __restrict__ b2,
                         const float* __restrict__ cb0,
                         const float* __restrict__ cb1,
                         const float* __restrict__ cb2,
                         float* __restrict__ out)
{
    // B-matrix staging: [W0|W1|W2|0] padded to 16 cols, interleaved by k-pairs
    // so each lane fetches its two K values as a single 8-byte LDS read.
    // Wlds[p][n] = { Wcat[2p][n], Wcat[2p+1][n] }   (64 KB)
    __shared__ v2f Wlds[E_ / 2][16];

    const int tid = threadIdx.x;
    for (int idx = tid; idx < E_ * 16; idx += 256) {
        const int k = idx >> 4;
        const int n = idx & 15;
        float w;
        if      (n < 8)  w = W0[k * 8 + n];
        else if (n < 12) w = W1[k * 4 + (n - 8)];
        else if (n < 14) w = W2[k * 2 + (n - 12)];
        else             w = 0.0f;
        ((float*)&Wlds[k >> 1][n])[k & 1] = w;
    }
    __syncthreads();

    const int lane = tid & 31;
    const int wave = tid >> 5;
    const int half = lane >> 4;   // lane half selects K sub-pair (A/B) / M group (C)
    const int n    = lane & 15;   // A: token-in-tile (M). B/C: column (N).
    const int tileBase = blockIdx.x * 128 + wave * 16;

    // ---- logits tile: D(16x16) += A(16x4) x B(4x16), K = 1024 in steps of 4 ----
    const float* arow = pt + (size_t)(tileBase + n) * E_ + half * 2;
    v8f acc = {0.f, 0.f, 0.f, 0.f, 0.f, 0.f, 0.f, 0.f};
    #pragma unroll 8
    for (int k = 0; k < E_; k += 4) {
        v2f a = *(const v2f*)(arow + k);            // global_load_b64 (A stream)
        v2f b = Wlds[(k >> 1) + half][n];           // ds_load_b64, conflict-free
        acc = __builtin_amdgcn_wmma_f32_16x16x4_f32(
                  false, a, false, b, (short)0, acc, false, false);
    }

    // ---- per-column static head metadata ----
    int head, col;
    if      (n < 8)  { head = 0; col = n;      }
    else if (n < 12) { head = 1; col = n - 8;  }
    else if (n < 14) { head = 2; col = n - 12; }
    else             { head = 3; col = 0;      }   // padding columns
    const float bias = (head == 0) ? b0[col]
                     : (head == 1) ? b1[col]
                     : (head == 2) ? b2[col] : 0.0f;
    // partner lane's head per xor distance (static: partner n = n ^ (1<<i))
    int ph[4];
    #pragma unroll
    for (int i = 0; i < 4; ++i) {
        const int on = n ^ (1 << i);
        ph[i] = (on < 8) ? 0 : (on < 12) ? 1 : (on < 14) ? 2 : 3;
    }

    const size_t BND = (size_t)B_ * N_ * D_;
    float4* __restrict__ o0 = (float4*)(out);
    float4* __restrict__ o1 = (float4*)(out + BND);
    float4* __restrict__ o2 = (float4*)(out + 2 * BND);
    const float4* __restrict__ cb0v = (const float4*)cb0;  // row = 16 float4
    const float4* __restrict__ cb1v = (const float4*)cb1;  // row = 64 float4
    const float4* __restrict__ cb2v = (const float4*)cb2;  // row = 128 float4

    #pragma unroll
    for (int v = 0; v < 8; ++v) {
        const int tok = tileBase + v + half * 8;   // token for this C VGPR row

        // perturbed logit for (tok, head, col); argmax(softmax) == argmax(l+g)
        float val;
        if      (head == 0) val = acc[v] + bias + gumbel_from_u(g0[(size_t)tok * 8 + col]);
        else if (head == 1) val = acc[v] + bias + gumbel_from_u(g1[(size_t)tok * 4 + col]);
        else if (head == 2) val = acc[v] + bias + gumbel_from_u(g2[(size_t)tok * 2 + col]);
        else                val = -3.0e38f;

        // segmented argmax inside the 16-lane half (heads occupy xor-closed spans)
        float best = val;
        int   bidx = col;
        #pragma unroll
        for (int i = 0; i < 4; ++i) {
            const float ov = __shfl_xor(best, 1 << i, 32);
            const int   oi = __shfl_xor(bidx, 1 << i, 32);
            if (ph[i] == head &&
                (ov > best || (ov == best && oi < bidx))) { best = ov; bidx = oi; }
        }

        // gather winners of this token's three heads from columns 0 / 8 / 12
        const int lb = half << 4;
        const int h0 = __shfl(bidx, lb + 0,  32);
        const int h1 = __shfl(bidx, lb + 8,  32);
        const int h2 = __shfl(bidx, lb + 12, 32);

        // hierarchical codebook gather: 3 x 64 floats, one float4 per lane
        const size_t o = (size_t)tok * 16 + n;
        o0[o] = cb0v[h0 * 16 + n];
        o1[o] = cb1v[h0 * 64 + h1 * 16 + n];
        o2[o] = cb2v[h0 * 128 + (h1 * 2 + h2) * 16 + n];
    }
}

extern "C" void kernel_launch(void* const* d_in, const int* in_sizes, int n_in,
                              void* d_out, int out_size, void* d_ws, size_t ws_size,
                              hipStream_t stream) {
    (void)in_sizes; (void)n_in; (void)out_size; (void)d_ws; (void)ws_size;
    const float* pt  = (const float*)d_in[0];
    const float* g0  = (const float*)d_in[1];
    const float* g1  = (const float*)d_in[2];
    const float* g2  = (const float*)d_in[3];
    const float* W0  = (const float*)d_in[4];
    const float* b0  = (const float*)d_in[5];
    const float* W1  = (const float*)d_in[6];
    const float* b1  = (const float*)d_in[7];
    const float* W2  = (const float*)d_in[8];
    const float* b2  = (const float*)d_in[9];
    const float* cb0 = (const float*)d_in[10];
    const float* cb1 = (const float*)d_in[11];
    const float* cb2 = (const float*)d_in[12];
    float* out = (float*)d_out;

    const int tokens = B_ * N_;          // 65536
    const int grid   = tokens / 128;     // 128 tokens per 256-thread block
    sae_quantize_kernel<<<grid, 256, 0, stream>>>(
        pt, g0, g1, g2, W0, b0, W1, b1, W2, b2, cb0, cb1, cb2, out);
}